// GaussianMixtureModeling_80058190397782
// MI455X (gfx1250) — compile-verified
//
#include <hip/hip_runtime.h>
#include <hip/hip_bf16.h>
#include <stddef.h>

// GMM EM, 8 iterations. B=262144, L=26 (xext=52 [+1 ones col]), K=64.
// E-step GEMM and M-step GEMM both via V_WMMA_F32_16X16X4_F32.

typedef float v2f __attribute__((ext_vector_type(2)));
typedef float v8f __attribute__((ext_vector_type(8)));

#define GMM_B      262144
#define GMM_L      26
#define GMM_LE     52            // [x, x*x]
#define GMM_K      64
#define GMM_TILES  (GMM_B / 16)  // 16384
#define LOG_PI_TRM 47.78480372664298f   // 26 * log(2*pi)

// workspace float offsets (contiguous):
// W[64] MU[1664] SIG[1664] CC[64] BM[52*64] Z[64] S1[64*52] LL[1]
#define OFF_W    0
#define OFF_MU   64
#define OFF_SIG  1728
#define OFF_CC   3392
#define OFF_BM   3456
#define OFF_Z    6784
#define OFF_S1   6848
#define OFF_LL   10176
#define ACC_LEN  3393            // Z + S1 + LL contiguous

// ---------------------------------------------------------------------------
// Main fused E+M pass. grid = 256 blocks x 256 threads (8 waves/block).
// Each wave independently processes 16-row tiles (grid-stride), no barriers.
// ---------------------------------------------------------------------------
__global__ void __launch_bounds__(256)
gmm_em_pass(const float* __restrict__ x, float* ws) {
  const float* CC = ws + OFF_CC;
  const float* BM = ws + OFF_BM;   // BM[l][k], l in [0,52), k in [0,64)
  float* Z  = ws + OFF_Z;
  float* S1 = ws + OFF_S1;
  float* LL = ws + OFF_LL;

  __shared__ float sXe[8][16 * 53];   // per-wave [x, x^2, 1] tile, stride 53
  __shared__ float sP [8][16 * 66];   // per-wave posterior tile, stride 66

  const int tid  = threadIdx.x;
  const int w    = tid >> 5;
  const int lane = tid & 31;
  const int ln   = lane & 15;
  const int half = lane >> 4;
  const int koff = half * 2;          // A/B fragment K-offset for upper half-wave

  float* xe = sXe[w];
  float* pp = sP[w];

  // ones column (feature 52) -> M-step GEMM row 52 produces z[k] for free
  if (lane < 16) xe[lane * 53 + GMM_LE] = 1.0f;

  float cct[4];
#pragma unroll
  for (int t = 0; t < 4; ++t) cct[t] = CC[16 * t + ln];

  float llv = 0.f;
  // M-step accumulators: D(64x64) = sum_tiles xe^T(64x16) @ p(16x64)
  // tile (mt,nt): rows 16mt..16mt+15 (features, row 52 = z), cols 16nt..16nt+15 (components)
  v8f macc[16] = {};

  const int gw = blockIdx.x * 8 + w;
  const int nw = gridDim.x * 8;

  for (int tile = gw; tile < GMM_TILES; tile += nw) {
    // ---- stage [x, x^2] for 16 rows into private LDS slice ----
    for (int idx = lane; idx < 16 * GMM_L; idx += 32) {
      int r = idx / GMM_L, l = idx - r * GMM_L;
      float v = x[(size_t)(tile * 16 + r) * GMM_L + l];
      xe[r * 53 + l]          = v;
      xe[r * 53 + GMM_L + l]  = v * v;
    }

    // ---- E-step GEMM: numer(16x64) = xe(16x52) @ BM(52x64), k-dim steps of 4 ----
    v8f acc[4] = {};
#pragma unroll
    for (int c = 0; c < 13; ++c) {
      v2f a;
      a[0] = xe[ln * 53 + 4 * c + koff];
      a[1] = xe[ln * 53 + 4 * c + koff + 1];
#pragma unroll
      for (int t = 0; t < 4; ++t) {
        v2f b;
        b[0] = BM[(4 * c + koff)     * GMM_K + 16 * t + ln];
        b[1] = BM[(4 * c + koff + 1) * GMM_K + 16 * t + ln];
        acc[t] = __builtin_amdgcn_wmma_f32_16x16x4_f32(
            false, a, false, b, (short)0, acc[t], false, false);
      }
    }
#pragma unroll
    for (int t = 0; t < 4; ++t)
#pragma unroll
      for (int j = 0; j < 8; ++j) acc[t][j] += cct[t];

    // ---- rowwise logsumexp + posterior (rows live in VGPR slot + half) ----
#pragma unroll
    for (int j = 0; j < 8; ++j) {
      float v0 = acc[0][j], v1 = acc[1][j], v2 = acc[2][j], v3 = acc[3][j];
      float m = fmaxf(fmaxf(v0, v1), fmaxf(v2, v3));
      m = fmaxf(m, __shfl_xor(m, 1, 32));
      m = fmaxf(m, __shfl_xor(m, 2, 32));
      m = fmaxf(m, __shfl_xor(m, 4, 32));
      m = fmaxf(m, __shfl_xor(m, 8, 32));
      float e0 = __expf(v0 - m), e1 = __expf(v1 - m);
      float e2 = __expf(v2 - m), e3 = __expf(v3 - m);
      float s = (e0 + e1) + (e2 + e3);
      s += __shfl_xor(s, 1, 32);
      s += __shfl_xor(s, 2, 32);
      s += __shfl_xor(s, 4, 32);
      s += __shfl_xor(s, 8, 32);
      float inv = 1.f / s;
      if (ln == 0) llv += m + __logf(s);   // one count per row (lanes 0 and 16)
      int row = j + half * 8;
      pp[row * 66 +      ln] = e0 * inv;
      pp[row * 66 + 16 + ln] = e1 * inv;
      pp[row * 66 + 32 + ln] = e2 * inv;
      pp[row * 66 + 48 + ln] = e3 * inv;
    }

    // ---- M-step GEMM: macc(64x64) += xe^T(64x16) @ p(16x64) ----
    // A[m][k] = xe[row k][feature m]  (rows m>=53 read stale/pad data: discarded)
    // B[k][n] = p[row k][component n]
#pragma unroll
    for (int rs = 0; rs < 4; ++rs) {
      const int k0 = 4 * rs + koff;
      v2f af[4], bf[4];
#pragma unroll
      for (int mt = 0; mt < 4; ++mt) {
        af[mt][0] = xe[k0       * 53 + 16 * mt + ln];
        af[mt][1] = xe[(k0 + 1) * 53 + 16 * mt + ln];
      }
#pragma unroll
      for (int nt = 0; nt < 4; ++nt) {
        bf[nt][0] = pp[k0       * 66 + 16 * nt + ln];
        bf[nt][1] = pp[(k0 + 1) * 66 + 16 * nt + ln];
      }
#pragma unroll
      for (int mt = 0; mt < 4; ++mt)
#pragma unroll
        for (int nt = 0; nt < 4; ++nt)
          macc[mt * 4 + nt] = __builtin_amdgcn_wmma_f32_16x16x4_f32(
              false, af[mt], false, bf[nt], (short)0, macc[mt * 4 + nt],
              false, false);
    }
  }

  // ---- global reduction: D rows 0..51 -> S1, row 52 -> Z, rows 53..63 junk ----
#pragma unroll
  for (int mt = 0; mt < 4; ++mt) {
#pragma unroll
    for (int nt = 0; nt < 4; ++nt) {
#pragma unroll
      for (int j = 0; j < 8; ++j) {
        int l = 16 * mt + j + 8 * half;
        int k = 16 * nt + ln;
        float v = macc[mt * 4 + nt][j];
        if (l < GMM_LE) {
          atomicAdd(&S1[k * GMM_LE + l], v);
        } else if (l == GMM_LE) {
          atomicAdd(&Z[k], v);
        }
      }
    }
  }
  llv += __shfl_xor(llv, 16, 32);
  llv += __shfl_xor(llv, 8, 32);
  llv += __shfl_xor(llv, 4, 32);
  llv += __shfl_xor(llv, 2, 32);
  llv += __shfl_xor(llv, 1, 32);
  if (lane == 0) atomicAdd(LL, llv);
}

// ---------------------------------------------------------------------------
// prep: single block. mode 0 = init from inputs; mode 1 = M-step closed form +
// rebuild params + zero accumulators; mode 2 = M-step closed form + write outputs.
// ---------------------------------------------------------------------------
__global__ void __launch_bounds__(256)
gmm_prep(const float* __restrict__ w0, const float* __restrict__ mu0,
         const float* __restrict__ sg0, float* ws, float* out, int mode) {
  float* W   = ws + OFF_W;
  float* MU  = ws + OFF_MU;
  float* SIG = ws + OFF_SIG;
  float* CC  = ws + OFF_CC;
  float* BM  = ws + OFF_BM;
  float* Z   = ws + OFF_Z;
  float* S1  = ws + OFF_S1;
  float* LL  = ws + OFF_LL;

  const int tid = threadIdx.x;
  __shared__ float sred[GMM_K];
  __shared__ float sSum;

  if (mode == 0) {
    if (tid < GMM_K) sred[tid] = w0[tid];
    __syncthreads();
    if (tid == 0) {
      float s = 0.f;
      for (int k = 0; k < GMM_K; ++k) s += sred[k];
      sSum = s;
    }
    __syncthreads();
    if (tid < GMM_K) W[tid] = w0[tid] / sSum;
    for (int idx = tid; idx < GMM_K * GMM_L; idx += blockDim.x) {
      MU[idx]  = mu0[idx];
      SIG[idx] = sg0[idx];
    }
  } else {
    float wn = 0.f;
    if (tid < GMM_K) {
      wn = fmaxf(Z[tid] * (1.f / (float)GMM_B), 1e-5f);
      sred[tid] = wn;
    }
    __syncthreads();
    if (tid == 0) {
      float s = 0.f;
      for (int k = 0; k < GMM_K; ++k) s += sred[k];
      sSum = s;
    }
    __syncthreads();
    if (tid < GMM_K) {
      const float sf = 1e-5f * (float)GMM_K;
      float a = (1.f - sf) / (sSum - sf);
      float b = 1e-5f * (1.f - a);
      W[tid] = a * wn + b;
    }
    for (int idx = tid; idx < GMM_K * GMM_L; idx += blockDim.x) {
      int k = idx / GMM_L, l = idx - k * GMM_L;
      float zinv = 1.f / Z[k];
      float mu = S1[k * GMM_LE + l] * zinv;
      float sg = fmaxf(S1[k * GMM_LE + GMM_L + l] * zinv - mu * mu, 1e-6f);
      MU[idx]  = mu;
      SIG[idx] = sg;
    }
  }
  __syncthreads();

  if (mode == 2) {
    // outputs: w[64], mu[64*26], sigma[64*26*26] (diag embed), ll[1]
    if (tid < GMM_K) out[tid] = W[tid];
    for (int idx = tid; idx < GMM_K * GMM_L; idx += blockDim.x)
      out[GMM_K + idx] = MU[idx];
    const int SB = GMM_K + GMM_K * GMM_L;               // 1728
    for (int idx = tid; idx < GMM_K * GMM_L * GMM_L; idx += blockDim.x) {
      int k = idx / (GMM_L * GMM_L);
      int rem = idx - k * (GMM_L * GMM_L);
      int r = rem / GMM_L, c = rem - r * GMM_L;
      out[SB + idx] = (r == c) ? SIG[k * GMM_L + r] : 0.f;
    }
    if (tid == 0) out[SB + GMM_K * GMM_L * GMM_L] = LL[0];   // index 44992
    return;
  }

  // rebuild per-component parameters for the next pass
  if (tid < GMM_K) {
    int k = tid;
    float logdet = 0.f, mmp = 0.f;
    for (int l = 0; l < GMM_L; ++l) {
      float s = SIG[k * GMM_L + l];
      float m = MU[k * GMM_L + l];
      logdet += __logf(s);
      mmp += m * m / s;
    }
    CC[k] = __logf(W[k]) - 0.5f * (LOG_PI_TRM + logdet + mmp);
  }
  for (int idx = tid; idx < GMM_K * GMM_L; idx += blockDim.x) {
    int k = idx / GMM_L, l = idx - k * GMM_L;
    float s = SIG[k * GMM_L + l];
    BM[l * GMM_K + k]           = MU[k * GMM_L + l] / s;   // mu * prec
    BM[(GMM_L + l) * GMM_K + k] = -0.5f / s;               // -0.5 * prec
  }
  // zero accumulators for the next pass
  for (int idx = tid; idx < ACC_LEN; idx += blockDim.x) Z[idx] = 0.f;
}

// ---------------------------------------------------------------------------
extern "C" void kernel_launch(void* const* d_in, const int* in_sizes, int n_in,
                              void* d_out, int out_size, void* d_ws, size_t ws_size,
                              hipStream_t stream) {
  (void)in_sizes; (void)n_in; (void)out_size; (void)ws_size;
  const float* x   = (const float*)d_in[0];
  const float* w0  = (const float*)d_in[1];
  const float* mu0 = (const float*)d_in[2];
  const float* sg0 = (const float*)d_in[3];
  float* out = (float*)d_out;
  float* ws  = (float*)d_ws;

  gmm_prep<<<1, 256, 0, stream>>>(w0, mu0, sg0, ws, out, 0);
  for (int it = 0; it < 8; ++it) {
    gmm_em_pass<<<256, 256, 0, stream>>>(x, ws);
    gmm_prep<<<1, 256, 0, stream>>>(w0, mu0, sg0, ws, out, (it == 7) ? 2 : 1);
  }
}